// OpinionMiner_51127290691598
// MI455X (gfx1250) — compile-verified
//
#include <hip/hip_runtime.h>
#include <hip/hip_bf16.h>
#include <math.h>

// Problem constants (match reference)
#define BB 32
#define SS 128
#define HH 768
#define MM 8      // max span width
#define MHh 6
#define KK 4      // KH == KT == 4
#define NSPAN (MM * SS)        // 1024 spans per sentence
#define NROW  (BB * SS)        // 4096 rows of the scoring GEMM
#define KITER (HH / 4)         // 192 WMMA k-steps

typedef float v2f __attribute__((ext_vector_type(2)));
typedef float v8f __attribute__((ext_vector_type(8)));

// ---------------------------------------------------------------------------
// Kernel 1: pack B operand of the WMMA GEMM into lane layout.
// B tile is 4x16 (KxN): VGPR r, lanes 0-15 -> row K=r, lanes 16-31 -> row K=r+2,
// N = lane & 15.  Column 0 = wh, column 1 = wt, rest zero.
// Bbuf[it*64 + r*32 + lane]
// ---------------------------------------------------------------------------
__global__ void om_pack_b(const float* __restrict__ wh, const float* __restrict__ wt,
                          float* __restrict__ Bbuf) {
    int i = blockIdx.x * blockDim.x + threadIdx.x;
    if (i >= KITER * 64) return;
    int it   = i >> 6;
    int r    = (i >> 5) & 1;
    int lane = i & 31;
    int K = r + ((lane >= 16) ? 2 : 0);
    int N = lane & 15;
    int k = it * 4 + K;
    float v = 0.f;
    if (N == 0)      v = wh[k];
    else if (N == 1) v = wt[k];
    Bbuf[i] = v;
}

// ---------------------------------------------------------------------------
// Kernel 2: scoring GEMM via V_WMMA_F32_16X16X4_F32.
// One wave handles 16 rows; 256 waves cover the 4096 rows.
// D[M][0] = dh[row], D[M][1] = dt[row].
// ---------------------------------------------------------------------------
__global__ __launch_bounds__(256) void om_gemm_scores(
        const float* __restrict__ E, const float* __restrict__ Bbuf,
        float* __restrict__ dh, float* __restrict__ dt) {
    const int lane = threadIdx.x & 31;
    const int wave = blockIdx.x * (blockDim.x >> 5) + (threadIdx.x >> 5);
    const int row0 = wave * 16;
    const int half = lane >> 4;          // 0 -> K {0,1}, 1 -> K {2,3}
    const int m    = lane & 15;          // A-matrix row within tile
    const float* aptr = E + (size_t)(row0 + m) * HH + half * 2;

    v8f acc = {};
    #pragma unroll 4
    for (int it = 0; it < KITER; ++it) {
        // prefetch the A stream ~4 cachelines ahead (speculative; OOB dropped)
        __builtin_prefetch(aptr + (it + 32) * 4, 0, 0);
        v2f a = *(const v2f*)(aptr + it * 4);        // A[M][kbase + half*2 + {0,1}]
        v2f b;
        b.x = Bbuf[it * 64 + lane];                  // K = half*2 + 0 row
        b.y = Bbuf[it * 64 + 32 + lane];             // K = half*2 + 1 row
        acc = __builtin_amdgcn_wmma_f32_16x16x4_f32(
                false, a, false, b, (short)0, acc, false, false);
    }
    // Extract D: lane (M>=8?16:0)+N holds D[M][N] in acc[M&7]; we need N=0,1.
    int n = lane & 15;
    if (n < 2) {
        float* outp = (n == 0) ? dh : dt;
        int mbase = half * 8;
        #pragma unroll
        for (int v = 0; v < 8; ++v) outp[row0 + mbase + v] = acc[v];
    }
}

// ---------------------------------------------------------------------------
// Kernel 3: per-batch scores + stable top-4.  One block (128 threads) per batch.
// ---------------------------------------------------------------------------
__global__ __launch_bounds__(128) void om_score_topk(
        const float* __restrict__ dh, const float* __restrict__ dt,
        const int* __restrict__ mask,
        const float* __restrict__ pbh, const float* __restrict__ pbt,
        float* __restrict__ hs_out, float* __restrict__ ts_out,
        int* __restrict__ tk_it, float* __restrict__ tk_vt,
        int* __restrict__ tk_ih, float* __restrict__ tk_vh) {
    __shared__ float s_dh[SS], s_dt[SS];
    __shared__ float s_pdh[SS + 1], s_pdt[SS + 1];
    __shared__ float s_hs[NSPAN], s_ts[NSPAN];
    __shared__ int   s_len;
    const int b = blockIdx.x;
    const int t = threadIdx.x;

    s_dh[t] = dh[b * SS + t];
    s_dt[t] = dt[b * SS + t];
    __syncthreads();

    if (t == 0) {
        float a = 0.f; s_pdh[0] = 0.f;
        for (int s = 0; s < SS; ++s) { a += s_dh[s]; s_pdh[s + 1] = a; }
    } else if (t == 1) {
        float a = 0.f; s_pdt[0] = 0.f;
        for (int s = 0; s < SS; ++s) { a += s_dt[s]; s_pdt[s + 1] = a; }
    } else if (t == 2) {
        int l = 0;
        for (int s = 0; s < SS; ++s) l += mask[b * SS + s];
        s_len = l;
    }
    __syncthreads();

    const float bh = *pbh, bt = *pbt;
    const int len = s_len;
    for (int rep = 0; rep < NSPAN / SS; ++rep) {
        int j    = rep * SS + t;
        int widx = j >> 7;
        int s    = j & (SS - 1);
        int w    = widx + 1;
        bool base = s < (len - widx);          // span fits inside sentence
        int e = s + w; if (e > SS) e = SS;     // clamp (only invalid spans exceed)
        float invw = 1.f / (float)w;
        float mh = (s_pdh[e] - s_pdh[s]) * invw;
        float mt = (s_pdt[e] - s_pdt[s]) * invw;
        float hsv = (base && (widx < MHh)) ? (1.f / (1.f + expf(-(mh + bh)))) : -1.0f;
        float tsv = base                   ? (1.f / (1.f + expf(-(mt + bt)))) : -1.0f;
        hs_out[b * NSPAN + j] = hsv;
        ts_out[b * NSPAN + j] = tsv;
        s_hs[j] = hsv;
        s_ts[j] = tsv;
    }
    __syncthreads();

    // Stable top-4 (descending, ties keep lower index first — matches lax.top_k)
    if (t < 2) {
        const float* buf = (t == 0) ? s_ts : s_hs;
        float v0 = -1e30f, v1 = -1e30f, v2 = -1e30f, v3 = -1e30f;
        int   i0 = 0, i1 = 0, i2 = 0, i3 = 0;
        for (int j = 0; j < NSPAN; ++j) {
            float v = buf[j];
            if (v > v3) {
                if (v > v0)      { v3=v2;i3=i2; v2=v1;i2=i1; v1=v0;i1=i0; v0=v;i0=j; }
                else if (v > v1) { v3=v2;i3=i2; v2=v1;i2=i1; v1=v;i1=j; }
                else if (v > v2) { v3=v2;i3=i2; v2=v;i2=j; }
                else             { v3=v;i3=j; }
            }
        }
        if (t == 0) {
            tk_vt[b*4+0]=v0; tk_vt[b*4+1]=v1; tk_vt[b*4+2]=v2; tk_vt[b*4+3]=v3;
            tk_it[b*4+0]=i0; tk_it[b*4+1]=i1; tk_it[b*4+2]=i2; tk_it[b*4+3]=i3;
        } else {
            tk_vh[b*4+0]=v0; tk_vh[b*4+1]=v1; tk_vh[b*4+2]=v2; tk_vh[b*4+3]=v3;
            tk_ih[b*4+0]=i0; tk_ih[b*4+1]=i1; tk_ih[b*4+2]=i2; tk_ih[b*4+3]=i3;
        }
    }
}

// ---------------------------------------------------------------------------
// Kernel 4: gather top-K span reps (recomputed from embedding, L2-hot) and
// produce implicit / explicit logits.  One block (8 waves) per batch:
// waves 0-3 -> target spans, waves 4-7 -> holder spans.
// ---------------------------------------------------------------------------
__global__ __launch_bounds__(256) void om_logits(
        const float* __restrict__ E,
        const float* __restrict__ wi, const float* __restrict__ bi,
        const float* __restrict__ we, const float* __restrict__ be,
        const int* __restrict__ tk_it, const float* __restrict__ tk_vt,
        const int* __restrict__ tk_ih, const float* __restrict__ tk_vh,
        float* __restrict__ implicit_out, float* __restrict__ explicit_out) {
    __shared__ float s_tI[KK][4];   // trep @ wi
    __shared__ float s_tE[KK][4];   // trep @ we[768:1536]
    __shared__ float s_hE[KK][4];   // hrep @ we[0:768]
    const int b    = blockIdx.x;
    const int t    = threadIdx.x;
    const int wv   = t >> 5;
    const int lane = t & 31;
    const bool isTarget = wv < KK;
    const int r = isTarget ? wv : wv - KK;

    const int   idx = isTarget ? tk_it[b*4+r] : tk_ih[b*4+r];
    const float val = isTarget ? tk_vt[b*4+r] : tk_vh[b*4+r];

    float a0=0,a1=0,a2=0,a3=0, e0=0,e1=0,e2=0,e3=0;
    if (val > 0.f) {                      // wave-uniform branch
        int widx = idx >> 7, s = idx & (SS - 1), w = widx + 1;
        const float* ebase = E + ((size_t)b * SS + s) * HH;
        const float* wev   = we + (isTarget ? HH * 4 : 0);
        for (int c = 0; c < HH / 32; ++c) {
            int h = c * 32 + lane;
            float v = 0.f;
            for (int u = 0; u < w; ++u) v += ebase[u * HH + h];
            float4 wr = *(const float4*)(wev + h * 4);
            e0 += v * wr.x; e1 += v * wr.y; e2 += v * wr.z; e3 += v * wr.w;
            if (isTarget) {
                float4 wir = *(const float4*)(wi + h * 4);
                a0 += v * wir.x; a1 += v * wir.y; a2 += v * wir.z; a3 += v * wir.w;
            }
        }
        float inv = 1.f / (float)w;
        for (int off = 16; off > 0; off >>= 1) {
            e0 += __shfl_down(e0, off, 32); e1 += __shfl_down(e1, off, 32);
            e2 += __shfl_down(e2, off, 32); e3 += __shfl_down(e3, off, 32);
            a0 += __shfl_down(a0, off, 32); a1 += __shfl_down(a1, off, 32);
            a2 += __shfl_down(a2, off, 32); a3 += __shfl_down(a3, off, 32);
        }
        if (lane == 0) {
            if (isTarget) {
                s_tI[r][0]=a0*inv; s_tI[r][1]=a1*inv; s_tI[r][2]=a2*inv; s_tI[r][3]=a3*inv;
                s_tE[r][0]=e0*inv; s_tE[r][1]=e1*inv; s_tE[r][2]=e2*inv; s_tE[r][3]=e3*inv;
            } else {
                s_hE[r][0]=e0*inv; s_hE[r][1]=e1*inv; s_hE[r][2]=e2*inv; s_hE[r][3]=e3*inv;
            }
        }
    } else if (lane == 0) {
        if (isTarget) { for (int c=0;c<4;++c){ s_tI[r][c]=0.f; s_tE[r][c]=0.f; } }
        else          { for (int c=0;c<4;++c){ s_hE[r][c]=0.f; } }
    }
    __syncthreads();

    if (t < KK * 4) {                     // implicit: [KT,4]
        int j = t >> 2, c = t & 3;
        float v = (tk_vt[b*4+j] > 0.f) ? (s_tI[j][c] + bi[c]) : bi[c];
        implicit_out[b * (KK*4) + t] = v;
    }
    if (t < KK * KK * 4) {                // explicit: [KH*KT,4], pair = i*KT+j
        int i = t >> 4, j = (t >> 2) & 3, c = t & 3;
        bool pv = (tk_vh[b*4+i] > 0.f) && (tk_vt[b*4+j] > 0.f);
        float v = pv ? (s_hE[i][c] + s_tE[j][c] + be[c]) : be[c];
        explicit_out[b * (KK*KK*4) + t] = v;
    }
}

// ---------------------------------------------------------------------------
// Launcher
// ---------------------------------------------------------------------------
extern "C" void kernel_launch(void* const* d_in, const int* in_sizes, int n_in,
                              void* d_out, int out_size, void* d_ws, size_t ws_size,
                              hipStream_t stream) {
    const float* E    = (const float*)d_in[0];   // [B,S,H]
    const int*   mask = (const int*)  d_in[1];   // [B,S]
    const float* wh   = (const float*)d_in[2];   // [H]
    const float* bh   = (const float*)d_in[3];   // scalar
    const float* wt   = (const float*)d_in[4];   // [H]
    const float* bt   = (const float*)d_in[5];   // scalar
    const float* wi   = (const float*)d_in[6];   // [H,4]
    const float* bi   = (const float*)d_in[7];   // [4]
    const float* we   = (const float*)d_in[8];   // [2H,4]
    const float* be   = (const float*)d_in[9];   // [4]

    // Output layout: implicit [32,4,4] | explicit [32,16,4] | hs [32,1024] | ts [32,1024]
    float* O            = (float*)d_out;
    float* implicit_out = O;
    float* explicit_out = O + BB * KK * 4;                       // +512
    float* hs_out       = explicit_out + BB * KK * KK * 4;       // +2048
    float* ts_out       = hs_out + BB * NSPAN;                   // +32768

    // Workspace layout (floats)
    float* W     = (float*)d_ws;
    float* Bbuf  = W;                       // 192*64 = 12288
    float* dh    = W + 12288;               // 4096
    float* dt    = W + 16384;               // 4096
    int*   tk_it = (int*)  (W + 20480);     // 128
    float* tk_vt =          W + 20608;      // 128
    int*   tk_ih = (int*)  (W + 20736);     // 128
    float* tk_vh =          W + 20864;      // 128

    om_pack_b<<<(KITER * 64 + 255) / 256, 256, 0, stream>>>(wh, wt, Bbuf);

    om_gemm_scores<<<NROW / 16 / 8, 256, 0, stream>>>(E, Bbuf, dh, dt);

    om_score_topk<<<BB, 128, 0, stream>>>(dh, dt, mask, bh, bt,
                                          hs_out, ts_out,
                                          tk_it, tk_vt, tk_ih, tk_vh);

    om_logits<<<BB, 256, 0, stream>>>(E, wi, bi, we, be,
                                      tk_it, tk_vt, tk_ih, tk_vh,
                                      implicit_out, explicit_out);
}